// NetSoNTopSIAMReg_20366734917782
// MI455X (gfx1250) — compile-verified
//
#include <hip/hip_runtime.h>

typedef float v2f __attribute__((ext_vector_type(2)));
typedef float v4f __attribute__((ext_vector_type(4)));
typedef float v8f __attribute__((ext_vector_type(8)));

#define NB   32          // batches
#define NA   102         // channels
#define PLN  50176       // 224*224
#define PLN4 12544       // PLN / 4
#define ROWP 104         // padded vote row (26 chunks of K=4)

// ---------------------------------------------------------------------------
// Stage 1: adaptive-avg-pool over 224x224 per (b,a) plane.
// One block per plane, 256 threads, exactly 49 float4 loads per thread.
// Non-temporal: 655 MB streamed once, don't pollute L2.
// ---------------------------------------------------------------------------
__global__ __launch_bounds__(256) void pool_mean_kernel(
    const float* __restrict__ maps, float* __restrict__ xsun)
{
    const int p   = blockIdx.x;            // plane = b*102 + a (contiguous)
    const int tid = threadIdx.x;
    const v4f* base = reinterpret_cast<const v4f*>(maps + (size_t)p * PLN);

    float s0 = 0.f, s1 = 0.f, s2 = 0.f, s3 = 0.f;
    #pragma unroll 7
    for (int i = 0; i < 49; ++i) {
        v4f v = __builtin_nontemporal_load(&base[i * 256 + tid]);
        s0 += v.x; s1 += v.y; s2 += v.z; s3 += v.w;
    }
    float s = (s0 + s1) + (s2 + s3);

    __shared__ float red[256];
    red[tid] = s;
    __syncthreads();
    #pragma unroll
    for (int off = 128; off > 0; off >>= 1) {
        if (tid < off) red[tid] += red[tid + off];
        __syncthreads();
    }
    if (tid == 0) xsun[p] = red[0] * (1.0f / (float)PLN);
}

// ---------------------------------------------------------------------------
// Stage 2: one wave32. vote = x_sun * weight; top-8 |.| prefix sums per batch;
// dense row-sums computed with V_WMMA_F32_16X16X4_F32 against an all-ones B.
// ---------------------------------------------------------------------------
__global__ __launch_bounds__(32) void topk_wmma_kernel(
    const float* __restrict__ xsun, const float* __restrict__ weight,
    const float* __restrict__ avgp, float* __restrict__ xson)
{
    __shared__ float vote[NB][ROWP];
    const int lane = threadIdx.x;          // 0..31; thread == batch row
    const float avg = avgp[0];

    // Build vote rows (zero-pad cols 102,103 so the WMMA K-loop is branchless)
    for (int a = 0; a < NA; ++a)
        vote[lane][a] = xsun[lane * NA + a] * weight[a];
    vote[lane][NA]     = 0.f;
    vote[lane][NA + 1] = 0.f;
    __syncthreads();

    // ---- dense pass: D = vote_tile(16xK) x ones(Kx16), accumulate K chunks.
    // A layout (16x4 f32): lanes 0-15 hold {K=0,K=1}, lanes 16-31 hold {K=2,K=3},
    // M = lane & 15. B = all-ones, so its layout is irrelevant.
    const int half = lane >> 4;
    const int m    = lane & 15;
    v2f bones; bones[0] = 1.0f; bones[1] = 1.0f;

    #pragma unroll
    for (int t = 0; t < 2; ++t) {          // two M-tiles of 16 batches
        v8f c = {};
        const int row = t * 16 + m;
        for (int kc = 0; kc < 26; ++kc) {
            const int col = kc * 4 + half * 2;
            v2f a;
            a[0] = vote[row][col];
            a[1] = vote[row][col + 1];
            c = __builtin_amdgcn_wmma_f32_16x16x4_f32(
                    false, a, false, bones, (short)0, c, false, false);
        }
        // Extract column N=0 of D: VGPR r, lane 0 -> M=r; lane 16 -> M=8+r.
        if (lane == 0) {
            #pragma unroll
            for (int r = 0; r < 8; ++r)
                xson[8 * NB + (t * 16 + r)] = c[r] + avg;
        } else if (lane == 16) {
            #pragma unroll
            for (int r = 0; r < 8; ++r)
                xson[8 * NB + (t * 16 + 8 + r)] = c[r] + avg;
        }
    }

    // ---- top-k: select 8 largest |vote| (stable on ties => first index wins,
    // matching argsort(-abs)); emit running prefix sums + avg.
    unsigned long long used0 = 0ull, used1 = 0ull;
    float run = 0.f;
    for (int k = 0; k < 8; ++k) {
        float bestAbs = -1.f, bestVal = 0.f;
        int bestIdx = 0;
        for (int a = 0; a < NA; ++a) {
            const bool taken = (a < 64) ? ((used0 >> a) & 1ull)
                                        : ((used1 >> (a - 64)) & 1ull);
            const float v  = vote[lane][a];
            const float av = fabsf(v);
            if (!taken && av > bestAbs) { bestAbs = av; bestVal = v; bestIdx = a; }
        }
        if (bestIdx < 64) used0 |= 1ull << bestIdx;
        else              used1 |= 1ull << (bestIdx - 64);
        run += bestVal;
        xson[k * NB + lane] = run + avg;   // x_son[k][b]
    }
}

// ---------------------------------------------------------------------------
extern "C" void kernel_launch(void* const* d_in, const int* in_sizes, int n_in,
                              void* d_out, int out_size, void* d_ws, size_t ws_size,
                              hipStream_t stream)
{
    const float* maps   = (const float*)d_in[0];   // [32,102,224,224]
    const float* weight = (const float*)d_in[1];   // [1,102]
    const float* avgv   = (const float*)d_in[2];   // scalar

    float* out  = (float*)d_out;
    float* xsun = out;                 // [32,102] = 3264 floats
    float* xson = out + NB * NA;       // [9,32]   = 288 floats

    pool_mean_kernel<<<dim3(NB * NA), dim3(256), 0, stream>>>(maps, xsun);
    topk_wmma_kernel<<<dim3(1), dim3(32), 0, stream>>>(xsun, weight, avgv, xson);
}